// Concat_Model_89343909692135
// MI455X (gfx1250) — compile-verified
//
#include <hip/hip_runtime.h>

typedef float v2f __attribute__((ext_vector_type(2)));
typedef float v4f __attribute__((ext_vector_type(4)));
typedef float v8f __attribute__((ext_vector_type(8)));

#define BATCH 2
#define NROW  4096
#define FDIM  320

// ---------------------------------------------------------------------------
// Stage 1: p_a = x1 . w_a, p_b = x1 . w_b   via V_WMMA_F32_16X16X4_F32
// One wave (32 lanes) per 16 rows. A = 16x4 tile of x1, B = 4x16 with
// column 0 = w_a, column 1 = w_b, other columns zero. Accumulate K=320 in
// 80 WMMA steps. D column 0 -> p_a, column 1 -> p_b.
// ---------------------------------------------------------------------------
__global__ __launch_bounds__(32) void dot_wmma_kernel(
    const float* __restrict__ x1,      // [B*N, F]
    const float* __restrict__ conv_w,  // [2*F]
    float* __restrict__ pa,            // [B*N]
    float* __restrict__ pb)            // [B*N]
{
    const int lane = threadIdx.x;        // 0..31
    const int n    = lane & 15;          // column / row-in-tile index
    const int g    = lane >> 4;          // K sub-group (0: K0..1, 1: K2..3)
    const int rowBase = blockIdx.x * 16; // 16 rows of x1 per wave

    // A-matrix: lane (g,n) supplies row M=n, K = k0 + 2*g + {0,1}
    const float* __restrict__ arow = x1 + (size_t)(rowBase + n) * FDIM + 2 * g;

    // B-matrix: column 0 = w_a, column 1 = w_b, columns 2..15 = 0
    const float* __restrict__ wA = conv_w;
    const float* __restrict__ wB = conv_w + FDIM;
    const float* __restrict__ wsel = (n == 1) ? wB : wA;
    const float msk = (n < 2) ? 1.0f : 0.0f;

    v8f c = {};
    for (int k0 = 0; k0 < FDIM; k0 += 4) {
        v2f a = *(const v2f*)(arow + k0);
        const int kk = k0 + 2 * g;
        v2f b;
        b.x = wsel[kk]     * msk;
        b.y = wsel[kk + 1] * msk;
        // D = A(16x4) * B(4x16) + C   (fp32 exact)
        c = __builtin_amdgcn_wmma_f32_16x16x4_f32(
                /*neg_a=*/false, a, /*neg_b=*/false, b,
                /*c_mod=*/(short)0, c, /*reuse_a=*/false, /*reuse_b=*/false);
    }

    // D layout: VGPR r, lanes 0-15 -> M=r, lanes 16-31 -> M=8+r; N = lane&15.
    // Column 0 (lanes 0,16) = p_a, column 1 (lanes 1,17) = p_b.
    if (n < 2) {
        float* __restrict__ dst = (n == 0) ? pa : pb;
        const int base = rowBase + 8 * g;
#pragma unroll
        for (int r = 0; r < 8; ++r) dst[base + r] = c[r];
    }
}

// ---------------------------------------------------------------------------
// Stage 2: out[b,i,j] = sigmoid(pb[b,i] + pa[b,j] + bias)
// Pure streaming-store kernel: 134 MB of b128 non-temporal stores is the
// roofline (~5.8 us @ 23.3 TB/s). Sigmoid via native v_exp_f32 / v_rcp_f32.
// One block per (b,i) row; 256 threads x 4 float4 each = 4096 columns.
// ---------------------------------------------------------------------------
__global__ __launch_bounds__(256) void outer_sigmoid_kernel(
    const float* __restrict__ pa,      // [B*N]
    const float* __restrict__ pb,      // [B*N]
    const float* __restrict__ conv_b,  // [1]
    float* __restrict__ out)           // [B*N, N]
{
    const int row  = blockIdx.x;             // b*N + i
    const int bIdx = row >> 12;              // row / 4096
    const float base = pb[row] + conv_b[0];  // uniform per block -> scalar regs

    const float* __restrict__ paRow = pa + (bIdx << 12);
    float* __restrict__ orow = out + (size_t)row * NROW;

    const int t = threadIdx.x;
#pragma unroll
    for (int cIt = 0; cIt < 4; ++cIt) {
        const int j = (t + (cIt << 8)) << 2;         // coalesced float4 lanes
        v4f pav = *(const v4f*)(paRow + j);
        v4f r;
#pragma unroll
        for (int e = 0; e < 4; ++e) {
            const float x  = base + pav[e];
            // sigmoid(x) = 1 / (1 + 2^(-x * log2(e)))
            const float ex = __builtin_amdgcn_exp2f(x * -1.44269504088896340736f);
            r[e] = __builtin_amdgcn_rcpf(1.0f + ex);
        }
        __builtin_nontemporal_store(r, (v4f*)(orow + j));
    }
}

// ---------------------------------------------------------------------------
extern "C" void kernel_launch(void* const* d_in, const int* in_sizes, int n_in,
                              void* d_out, int out_size, void* d_ws, size_t ws_size,
                              hipStream_t stream) {
    const float* x1     = (const float*)d_in[0];   // [2,4096,320] f32
    const float* conv_w = (const float*)d_in[1];   // [640] f32
    const float* conv_b = (const float*)d_in[2];   // [1] f32
    float* out = (float*)d_out;                    // [2,4096,4096] f32

    float* pa = (float*)d_ws;                      // [8192] f32
    float* pb = pa + BATCH * NROW;                 // [8192] f32 (64 KB total)

    // Stage 1: 8192 rows / 16 rows-per-wave = 512 single-wave blocks
    dot_wmma_kernel<<<(BATCH * NROW) / 16, 32, 0, stream>>>(x1, conv_w, pa, pb);

    // Stage 2: one block per output row
    outer_sigmoid_kernel<<<BATCH * NROW, 256, 0, stream>>>(pa, pb, conv_b, out);
}